// DiffPool_17755394802134
// MI455X (gfx1250) — compile-verified
//
#include <hip/hip_runtime.h>
#include <hip/hip_bf16.h>
#include <math.h>

#define NN 2048      // nodes
#define BB 8         // batch
#define FD 128       // features (= C = K_clusters)

typedef __attribute__((ext_vector_type(2))) float v2f;
typedef __attribute__((ext_vector_type(8))) float v8f;

__device__ __forceinline__ v8f wmma_f32_16x16x4(v2f a, v2f b, v8f c) {
    // D = A(16x4) * B(4x16) + C(16x16), fp32, wave32
    return __builtin_amdgcn_wmma_f32_16x16x4_f32(
        /*neg_a=*/false, a, /*neg_b=*/false, b,
        /*c_mod=*/(short)0, c, /*reuse_a=*/false, /*reuse_b=*/false);
}

__device__ __forceinline__ float wave_sum(float v) {
    #pragma unroll
    for (int off = 16; off > 0; off >>= 1) v += __shfl_xor(v, off, 32);
    return v;
}
__device__ __forceinline__ float wave_max(float v) {
    #pragma unroll
    for (int off = 16; off > 0; off >>= 1) v = fmaxf(v, __shfl_xor(v, off, 32));
    return v;
}

// ---------------- K0: zero loss accumulators (lp[8], entr[1]) ----------------
__global__ void k_init(float* __restrict__ acc) {
    if (threadIdx.x < 9) acc[threadIdx.x] = 0.0f;
}

// ---------------- K1: d = rsqrt(rowsum(A) + 1)  (one wave per row) ----------
__global__ void k_degree(const float* __restrict__ A, float* __restrict__ d) {
    int row  = blockIdx.x * (blockDim.x >> 5) + (threadIdx.x >> 5); // [0, B*N)
    int lane = threadIdx.x & 31;
    const float* Ar = A + (size_t)row * NN;   // A is [B*N, N] flat
    float s = 0.0f;
    for (int m = lane; m < NN; m += 32) s += Ar[m];
    s = wave_sum(s) + 1.0f;                   // +1: identity diagonal
    if (lane == 0) d[row] = rsqrtf(s);
}

// ---------------- K2: XW = X @ [W_emb | W_pool]  -> [B*N, 256] ---------------
__global__ void k_project(const float* __restrict__ X, const float* __restrict__ We,
                          const float* __restrict__ Wp, float* __restrict__ XW) {
    int wave = threadIdx.x >> 5, lane = threadIdx.x & 31;
    int tile = blockIdx.x * (blockDim.x >> 5) + wave;   // 1024 row-tiles x 8 col-pairs
    int m0 = (tile >> 3) * 16;
    int c0 = (tile & 7) * 16;                 // tile0 in W_emb cols, tile1 = +128 in W_pool
    int lm = lane & 15;
    int kh = (lane >> 4) << 1;                // 0 or 2 (K sub-step per half-wave)
    v8f acc0 = {}, acc1 = {};
    #pragma unroll
    for (int k = 0; k < FD; k += 4) {
        v2f a, b0, b1;
        const float* Xr = X + (size_t)(m0 + lm) * FD + k + kh;
        a.x = Xr[0]; a.y = Xr[1];
        const float* Wr0 = We + (size_t)(k + kh) * FD + c0 + lm;
        b0.x = Wr0[0]; b0.y = Wr0[FD];
        const float* Wr1 = Wp + (size_t)(k + kh) * FD + c0 + lm;
        b1.x = Wr1[0]; b1.y = Wr1[FD];
        acc0 = wmma_f32_16x16x4(a, b0, acc0); // shares A-fragment
        acc1 = wmma_f32_16x16x4(a, b1, acc1);
    }
    #pragma unroll
    for (int i = 0; i < 8; i++) {
        size_t r = (size_t)(m0 + i + ((lane >> 4) << 3)) * 256;
        XW[r + c0 + lm]        = acc0[i];
        XW[r + 128 + c0 + lm]  = acc1[i];
    }
}

// ------- K3/K5: Out = Mhat(A) @ Bmat ; Mhat = (A [+ I]) [* d_n d_m] ---------
// grid: x = row tile (N/16), z = batch; block = 256 (8 waves).
// CT = column tiles per wave (wave w owns cols w*16 + t*128, t<CT).
template <bool NORM, bool ADD_DIAG, int CT>
__global__ void k_prop(const float* __restrict__ A, const float* __restrict__ d,
                       const float* __restrict__ Bmat, int ldb,
                       float* __restrict__ Out, int ldo) {
    __shared__ float As[16 * 68];             // 16 rows x 64 K, padded stride
    int b    = blockIdx.z;
    int row0 = blockIdx.x * 16;
    int wave = threadIdx.x >> 5, lane = threadIdx.x & 31;
    int c0   = wave * 16;
    const float* Ab = A + (size_t)b * NN * NN;
    const float* Bb = Bmat + (size_t)b * NN * ldb;
    const float* db = d + b * NN;
    int lm = lane & 15, kh = (lane >> 4) << 1;
    v8f acc[CT];
    #pragma unroll
    for (int t = 0; t < CT; t++) acc[t] = (v8f){};
    for (int kk = 0; kk < NN; kk += 64) {
        __syncthreads();                      // protect LDS vs previous iter reads
        for (int e = threadIdx.x; e < 16 * 64; e += blockDim.x) {
            int r = e >> 6, c = e & 63;
            int gr = row0 + r, gc = kk + c;
            float v = Ab[(size_t)gr * NN + gc];
            if (ADD_DIAG && gr == gc) v += 1.0f;
            if (NORM) v *= db[gr] * db[gc];
            As[r * 68 + c] = v;
            // prefetch next K-slab of A (one probe per 64B line) -> global_prefetch_b8
            if ((kk + 64 < NN) && ((e & 15) == 0))
                __builtin_prefetch(&Ab[(size_t)gr * NN + gc + 64], 0, 1);
        }
        __syncthreads();
        #pragma unroll
        for (int k4 = 0; k4 < 64; k4 += 4) {
            v2f a;
            a.x = As[lm * 68 + k4 + kh];
            a.y = As[lm * 68 + k4 + kh + 1];
            #pragma unroll
            for (int t = 0; t < CT; t++) {    // A-fragment reused CT times
                v2f bf;
                const float* Br = Bb + (size_t)(kk + k4 + kh) * ldb + t * 128 + c0 + lm;
                bf.x = Br[0]; bf.y = Br[ldb];
                acc[t] = wmma_f32_16x16x4(a, bf, acc[t]);
            }
        }
    }
    float* Ob = Out + (size_t)b * NN * ldo;
    #pragma unroll
    for (int i = 0; i < 8; i++) {
        size_t r = (size_t)(row0 + i + ((lane >> 4) << 3)) * ldo;
        #pragma unroll
        for (int t = 0; t < CT; t++)
            Ob[r + t * 128 + c0 + lm] = acc[t][i];
    }
}

// -------- K4: softmax rows of P (= ZP cols 128..255) -> S, S^T + entropy ----
__global__ void k_softmax(const float* __restrict__ ZP, float* __restrict__ S,
                          float* __restrict__ ST, float* __restrict__ entr_acc) {
    int row  = blockIdx.x * (blockDim.x >> 5) + (threadIdx.x >> 5);
    int lane = threadIdx.x & 31;
    const float* Pr = ZP + (size_t)row * 256 + 128;
    float v[4];
    float mx = -3.402823466e38f;
    #pragma unroll
    for (int i = 0; i < 4; i++) { v[i] = Pr[lane + 32 * i]; mx = fmaxf(mx, v[i]); }
    mx = wave_max(mx);
    float s = 0.0f;
    #pragma unroll
    for (int i = 0; i < 4; i++) { v[i] = __expf(v[i] - mx); s += v[i]; }
    s = wave_sum(s);
    float inv = 1.0f / s;
    int b = row >> 11, n = row & (NN - 1);
    float* Sr  = S + (size_t)row * FD;
    float* STb = ST + (size_t)b * FD * NN;
    float ent = 0.0f;
    #pragma unroll
    for (int i = 0; i < 4; i++) {
        float sv = v[i] * inv;
        Sr[lane + 32 * i] = sv;
        STb[(size_t)(lane + 32 * i) * NN + n] = sv;   // transposed copy for k_pool
        ent -= sv * __logf(sv + 1e-7f);
    }
    ent = wave_sum(ent);
    if (lane == 0) atomicAdd(entr_acc, ent);
}

// ---------------- K6: sum (A - S S^T)^2 per batch via WMMA Gram tiles -------
// Each wave computes a 16x32 Gram region (2 tiles sharing the A-fragment).
__global__ void k_lploss(const float* __restrict__ A, const float* __restrict__ S,
                         float* __restrict__ lp_acc) {
    int wave = threadIdx.x >> 5, lane = threadIdx.x & 31;
    int b    = blockIdx.z;
    int tile = blockIdx.x * (blockDim.x >> 5) + wave;   // 128 n-tiles x 64 m-pairs
    int n0 = (tile >> 6) * 16, m0 = (tile & 63) * 32;
    const float* Sb = S + (size_t)b * NN * FD;
    int lm = lane & 15, kh = (lane >> 4) << 1;
    v8f acc0 = {}, acc1 = {};
    #pragma unroll
    for (int k = 0; k < FD; k += 4) {
        v2f a, b0, b1;
        const float* Sa = Sb + (size_t)(n0 + lm) * FD + k + kh;
        a.x = Sa[0]; a.y = Sa[1];
        const float* Sm0 = Sb + (size_t)(m0 + lm) * FD + k + kh;       // B[k][j]=S[m0+j][k]
        b0.x = Sm0[0]; b0.y = Sm0[1];
        const float* Sm1 = Sb + (size_t)(m0 + 16 + lm) * FD + k + kh;
        b1.x = Sm1[0]; b1.y = Sm1[1];
        acc0 = wmma_f32_16x16x4(a, b0, acc0);
        acc1 = wmma_f32_16x16x4(a, b1, acc1);
    }
    const float* Ab = A + (size_t)b * NN * NN;
    float partial = 0.0f;
    #pragma unroll
    for (int i = 0; i < 8; i++) {
        size_t r = (size_t)(n0 + i + ((lane >> 4) << 3)) * NN;
        float d0 = Ab[r + m0 + lm] - acc0[i];
        float d1 = Ab[r + m0 + 16 + lm] - acc1[i];
        partial += d0 * d0 + d1 * d1;
    }
    partial = wave_sum(partial);
    if (lane == 0) atomicAdd(&lp_acc[b], partial);
}

// ---------------- K7: Out[b] = S[b]^T @ Bmat[b]  (uses transposed S) --------
__global__ void k_pool(const float* __restrict__ ST, const float* __restrict__ Bmat,
                       int ldb, float* __restrict__ Out) {
    int wave = threadIdx.x >> 5, lane = threadIdx.x & 31;
    int tile = blockIdx.x * (blockDim.x >> 5) + wave;   // 64 tiles per batch
    int b    = blockIdx.z;
    int p0 = (tile >> 3) * 16, c0 = (tile & 7) * 16;
    const float* STb = ST + (size_t)b * FD * NN;
    const float* Bb  = Bmat + (size_t)b * NN * ldb;
    int lm = lane & 15, kh = (lane >> 4) << 1;
    v8f acc = {};
    #pragma unroll 8
    for (int n = 0; n < NN; n += 4) {
        v2f a, bf;
        const float* Sa = STb + (size_t)(p0 + lm) * NN + n + kh;  // contiguous b64 now
        a.x = Sa[0]; a.y = Sa[1];
        const float* Br = Bb + (size_t)(n + kh) * ldb + c0 + lm;
        bf.x = Br[0]; bf.y = Br[ldb];
        acc = wmma_f32_16x16x4(a, bf, acc);
    }
    float* Ob = Out + (size_t)b * FD * FD;
    #pragma unroll
    for (int i = 0; i < 8; i++)
        Ob[(size_t)(p0 + i + ((lane >> 4) << 3)) * FD + c0 + lm] = acc[i];
}

// ---------------- K8: finalize scalar losses --------------------------------
__global__ void k_finalize(const float* __restrict__ acc,
                           float* __restrict__ out_lp, float* __restrict__ out_ent) {
    if (threadIdx.x == 0) {
        float s = 0.0f;
        for (int b = 0; b < BB; b++) s += sqrtf(acc[b]);
        *out_lp  = s / (float)BB;
        *out_ent = acc[8] / (float)(BB * NN);
    }
}

extern "C" void kernel_launch(void* const* d_in, const int* in_sizes, int n_in,
                              void* d_out, int out_size, void* d_ws, size_t ws_size,
                              hipStream_t stream) {
    (void)in_sizes; (void)n_in; (void)out_size; (void)ws_size;
    const float* X  = (const float*)d_in[0];   // [8,2048,128]
    const float* A  = (const float*)d_in[1];   // [8,2048,2048]
    const float* We = (const float*)d_in[2];   // [128,128]
    const float* Wp = (const float*)d_in[3];   // [128,128]

    float* out = (float*)d_out;
    float* Xp  = out;                                   // [8,128,128]
    float* Ap  = out + (size_t)BB * FD * FD;            // [8,128,128]
    float* S   = out + (size_t)2 * BB * FD * FD;        // [8,2048,128]
    float* lp_out  = S + (size_t)BB * NN * FD;          // scalar
    float* ent_out = lp_out + 1;                        // scalar

    float* ws   = (float*)d_ws;
    float* dvec = ws;                                   // 16384
    float* XW   = dvec + (size_t)BB * NN;               // [16384, 256] = [Xe|Xp]
    float* ZP   = XW + (size_t)BB * NN * 256;           // [16384, 256] = [Z|P]
    float* AS   = ZP + (size_t)BB * NN * 256;           // [16384, 128]
    float* STt  = AS + (size_t)BB * NN * FD;            // [8, 128, 2048] = S^T
    float* acc  = STt + (size_t)BB * FD * NN;           // lp[8] + entr[1]

    k_init<<<1, 32, 0, stream>>>(acc);
    k_degree<<<(BB * NN) / 8, 256, 0, stream>>>(A, dvec);
    k_project<<<(BB * NN / 16) * 8 / 8, 256, 0, stream>>>(X, We, Wp, XW);

    dim3 gzp(NN / 16, 1, BB);   // fltr @ [XWe|XWp] -> [Z|P], 2 col tiles / wave
    k_prop<true, true, 2><<<gzp, 256, 0, stream>>>(A, dvec, XW, 256, ZP, 256);

    k_softmax<<<(BB * NN) / 8, 256, 0, stream>>>(ZP, S, STt, acc + 8);

    dim3 gas(NN / 16, 1, BB);   // A @ S -> AS (raw adjacency for A_pooled)
    k_prop<false, false, 1><<<gas, 256, 0, stream>>>(A, dvec, S, FD, AS, FD);

    dim3 glp((NN / 16) * (NN / 32) / 8, 1, BB);   // 2 Gram tiles / wave
    k_lploss<<<glp, 256, 0, stream>>>(A, S, acc);

    dim3 gp((FD / 16) * (FD / 16) / 8, 1, BB);
    k_pool<<<gp, 256, 0, stream>>>(STt, ZP, 256, Xp);   // X_pooled = S^T Z
    k_pool<<<gp, 256, 0, stream>>>(STt, AS, FD, Ap);    // A_pooled = S^T (A S)

    k_finalize<<<1, 32, 0, stream>>>(acc, lp_out, ent_out);
}